// ScaleLayer_1073741824554
// MI455X (gfx1250) — compile-verified
//
#include <hip/hip_runtime.h>

// ---------------------------------------------------------------------------
// Problem constants (from the reference): N=8, Jr=48, M=16, Jq=64, K=64,
// T=4096, P=512.  out[n,p,c] = sum_jr W[p,jr] * x[n,jr,c], c in [0, T*2).
// W[p,jr] = L[k_idx[p], idx_r[jr]] if mapping1[jr]==mapping2[p] and jr is the
// LAST jr writing slot (mapping1[jr], idx_r[jr])  (scatter .set semantics),
// else 0.
// ---------------------------------------------------------------------------
#define NN   8
#define JRr  48
#define PP   512
#define JQ   64
#define T2   8192          // T*2 contiguous fastest-moving floats
#define KB   12            // 48 / 4 k-blocks per 16x16 tile

typedef __attribute__((ext_vector_type(2))) float v2f;
typedef __attribute__((ext_vector_type(8))) float v8f;

// ---------------------------------------------------------------------------
// Phase 1: materialize the 512x48 effective weight matrix in d_ws (96 KB).
// ---------------------------------------------------------------------------
__global__ void build_w_kernel(const float* __restrict__ L,
                               const int*   __restrict__ mapping1,
                               const int*   __restrict__ idx_r,
                               const int*   __restrict__ mapping2,
                               const int*   __restrict__ k_idx,
                               float*       __restrict__ W) {
    int tid = blockIdx.x * blockDim.x + threadIdx.x;
    if (tid >= PP * JRr) return;
    int p  = tid / JRr;
    int jr = tid - p * JRr;

    int m = mapping1[jr];
    int q = idx_r[jr];

    // last-writer-wins: jr contributes only if no later jr' hits the same slot
    bool winner = true;
    for (int j2 = jr + 1; j2 < JRr; ++j2) {
        if (mapping1[j2] == m && idx_r[j2] == q) { winner = false; break; }
    }

    float w = 0.0f;
    if (winner && mapping2[p] == m) {
        w = L[k_idx[p] * JQ + q];
    }
    W[p * JRr + jr] = w;
}

// ---------------------------------------------------------------------------
// Phase 2: out_n[512 x 8192] = W[512 x 48] * x_n[48 x 8192] via
// V_WMMA_F32_16X16X4_F32 (fp32, K=4 -> 12 chained WMMAs per 16x16 tile).
//
// Wave decomposition: wave = (n, p_tile, column_chunk); 8*32*32 = 8192 waves.
// Each wave loads its 16x48 slice of W once (12 x v2f A operands) and sweeps
// 16 column tiles (256 output columns), keeping A resident in VGPRs.
//
// Operand layouts (ISA 7.12.2, 32-bit):
//   A 16x4 : lane<16  -> M=lane,    v0=K0, v1=K1
//            lane>=16 -> M=lane-16, v0=K2, v1=K3
//   B 4x16 : VGPRv, lane -> K = v + 2*(lane>>4), N = lane&15
//   D 16x16: VGPRv, lane -> M = v + 8*(lane>>4), N = lane&15
// ---------------------------------------------------------------------------
__global__ void __launch_bounds__(256)
gemm_wmma_kernel(const float* __restrict__ x,
                 const float* __restrict__ W,
                 float*       __restrict__ out) {
    const int lane  = threadIdx.x & 31;
    const int wave  = blockIdx.x * (blockDim.x >> 5) + (threadIdx.x >> 5);

    const int chunk = wave & 31;          // column chunk: 256 cols each
    const int ptile = (wave >> 5) & 31;   // 16-row tile of P
    const int n     = wave >> 10;         // batch 0..7

    const int half  = lane >> 4;          // 0 or 1 (lane group)
    const int lcol  = lane & 15;          // N index / M row within tile
    const int p0    = ptile * 16;

    // ---- Load A operands: W rows p0..p0+15, all 48 K values (24 VGPRs) ----
    v2f A[KB];
    {
        const float* wrow = W + (p0 + lcol) * JRr + half * 2;
        #pragma unroll
        for (int kb = 0; kb < KB; ++kb) {
            A[kb] = *(const v2f*)(wrow + kb * 4);   // (K0,K1) or (K2,K3)
        }
    }

    const float* xn    = x   + (size_t)n * JRr * T2;
    float*       on    = out + (size_t)n * PP  * T2;
    const int    cbase = chunk * 256;

    #pragma unroll 1
    for (int ct = 0; ct < 16; ++ct) {
        const int    c0 = cbase + ct * 16;
        const float* xc = xn + c0 + lcol;

        // Pull the next column tile toward L2/L0 while we compute this one.
        __builtin_prefetch(xc + 16, 0, 1);

        // Two accumulators to break the WMMA D->C dependency chain.
        v8f acc0 = {};
        v8f acc1 = {};
        #pragma unroll
        for (int kb = 0; kb < KB; kb += 2) {
            {
                const int jr = kb * 4 + half * 2;
                v2f b;
                b.x = xc[(size_t)jr * T2];          // K = 2*half
                b.y = xc[(size_t)(jr + 1) * T2];    // K = 2*half + 1
                acc0 = __builtin_amdgcn_wmma_f32_16x16x4_f32(
                           false, A[kb], false, b, (short)0, acc0, false, false);
            }
            {
                const int jr = (kb + 1) * 4 + half * 2;
                v2f b;
                b.x = xc[(size_t)jr * T2];
                b.y = xc[(size_t)(jr + 1) * T2];
                acc1 = __builtin_amdgcn_wmma_f32_16x16x4_f32(
                           false, A[kb + 1], false, b, (short)0, acc1, false, false);
            }
        }
        v8f acc = acc0 + acc1;

        // ---- Store D: VGPR v holds rows (v + 8*half), cols = lcol ----
        float* oc = on + (size_t)(p0 + half * 8) * T2 + c0 + lcol;
        #pragma unroll
        for (int v = 0; v < 8; ++v) {
            oc[(size_t)v * T2] = acc[v];
        }
    }
}

// ---------------------------------------------------------------------------
// Launch: inputs in setup_inputs() order:
//   0:x(f32) 1:L(f32) 2:mapping1 3:idx_r 4:mapping2 5:sums(unused)
//   6:k_idx 7:j_idx(unused)
// d_out: 8*512*8192 floats.  d_ws: 96 KB used for W.
// ---------------------------------------------------------------------------
extern "C" void kernel_launch(void* const* d_in, const int* in_sizes, int n_in,
                              void* d_out, int out_size, void* d_ws, size_t ws_size,
                              hipStream_t stream) {
    const float* x        = (const float*)d_in[0];
    const float* L        = (const float*)d_in[1];
    const int*   mapping1 = (const int*)d_in[2];
    const int*   idx_r    = (const int*)d_in[3];
    const int*   mapping2 = (const int*)d_in[4];
    const int*   k_idx    = (const int*)d_in[6];

    float* W = (float*)d_ws;

    build_w_kernel<<<(PP * JRr + 255) / 256, 256, 0, stream>>>(
        L, mapping1, idx_r, mapping2, k_idx, W);

    // 8192 waves = 1024 blocks * 8 waves: exactly N(8) * ptiles(32) * chunks(32)
    gemm_wmma_kernel<<<1024, 256, 0, stream>>>(x, W, (float*)d_out);
}